// gHHCTree_82282983456821
// MI455X (gfx1250) — compile-verified
//
#include <hip/hip_runtime.h>
#include <hip/hip_bf16.h>

#define GAMMA_C 0.25f
#define NN 16384
#define MM 2048
#define DD 64
#define MTILES (MM / 16)   // 128 column tiles
#define NWAVES 8
#define BLK 256
#define XPAD 68            // LDS row stride in floats (bank-conflict-free b64 A loads)

typedef float v2f __attribute__((ext_vector_type(2)));
typedef float v8f __attribute__((ext_vector_type(8)));

// ---------------------------------------------------------------- helpers

__device__ __forceinline__ float softplusf(float x) {
    // log1p(exp(x)) stably
    return fmaxf(x, 0.f) + log1pf(__expf(-fabsf(x)));
}

// Poincare distance * parent-order penalty for one (row, col) element
__device__ __forceinline__ float pdistf(float dot, float sqx, float pnx,
                                        float sqn_, float pnn_) {
    float num = sqx + sqn_ - 2.f * dot;          // squared euclid cdist
    float den = (1.f - sqx) * (1.f - sqn_);
    float u   = fmaxf(2.f * num / den, 0.f);     // arccosh(1+u) = log1p(u + sqrt(u(u+2)))
    float dist = log1pf(u + sqrtf(u * (u + 2.f)));
    float pen  = fmaxf(0.f, pnn_ - pnx + GAMMA_C) + 1.f;
    return dist * pen;
}

// Combine two online-softmax + argmin stat sets:
// (m1,s1): softmax stats for z1 ; (m2,s2): softmax stats for z2 ;
// (vm,im): running argmin of max_ij ; c2: exp(z2@argmin - m2) in local frame.
__device__ __forceinline__ void combine_stats(float& m1, float& s1,
                                              float& m2, float& s2,
                                              float& vm, float& c2, int& im,
                                              float om1, float os1,
                                              float om2, float os2,
                                              float ovm, float oc2, int oim) {
    float nm1 = fmaxf(m1, om1);
    s1 = s1 * __expf(m1 - nm1) + os1 * __expf(om1 - nm1);
    m1 = nm1;
    float nm2 = fmaxf(m2, om2);
    float myc = c2  * __expf(m2  - nm2);
    float otc = oc2 * __expf(om2 - nm2);
    s2 = s2 * __expf(m2 - nm2) + os2 * __expf(om2 - nm2);
    m2 = nm2;
    bool take = (ovm < vm) || ((ovm == vm) && (oim < im));  // first-occurrence tie-break
    vm = take ? ovm : vm;
    im = take ? oim : im;
    c2 = take ? otc : myc;
}

// ------------------------------------------------- per-row norm stats kernel
// one wave (32 lanes) per row of a [R,64] matrix: sq = ||x||^2, pn = 2*atanh(||x||)
__global__ __launch_bounds__(BLK) void rowstats_kernel(const float* __restrict__ x,
                                                       float* __restrict__ sq,
                                                       float* __restrict__ pn,
                                                       int R) {
    const int wave = threadIdx.x >> 5;
    const int lane = threadIdx.x & 31;
    const int row  = blockIdx.x * NWAVES + wave;
    if (row >= R) return;
    float a = x[(size_t)row * DD + lane];
    float b = x[(size_t)row * DD + lane + 32];
    float s = a * a + b * b;
#pragma unroll
    for (int off = 16; off >= 1; off >>= 1) s += __shfl_xor(s, off, 32);
    if (lane == 0) {
        sq[row] = s;
        pn[row] = 2.f * atanhf(sqrtf(s));
    }
}

// ---------------------------------------------------------------- main kernel
// One workgroup handles 16 consecutive rows (a full WMMA M-tile) of all three
// X matrices. 8 waves split the 128 column tiles. Two phases (stats, loss).
__global__ __launch_bounds__(BLK) void hhc_main_kernel(
    const float* __restrict__ xi, const float* __restrict__ xj,
    const float* __restrict__ xk, const float* __restrict__ nodes,
    const float* __restrict__ uij, const float* __restrict__ uijk,
    const float* __restrict__ sqn, const float* __restrict__ pnn,
    const float* __restrict__ sqxi, const float* __restrict__ pnxi,
    const float* __restrict__ sqxj, const float* __restrict__ pnxj,
    const float* __restrict__ sqxk, const float* __restrict__ pnxk,
    float* __restrict__ partials) {
    __shared__ float sX[3][16][XPAD];          // staged X row tiles
    __shared__ float sSq[3][16], sPn[3][16];   // per-row norm stats
    __shared__ float redM1[NWAVES][16], redS1[NWAVES][16];
    __shared__ float redM2[NWAVES][16], redS2[NWAVES][16];
    __shared__ float redVM[NWAVES][16], redC2[NWAVES][16];
    __shared__ int   redIM[NWAVES][16];
    __shared__ float rowM1[16], rowS1[16], rowM2[16], rowZp[16];
    __shared__ int   rowIm[16];
    __shared__ float waveLoss[NWAVES];

    const int tid  = threadIdx.x;
    const int wave = tid >> 5;
    const int lane = tid & 31;
    const int half = lane >> 4;     // 0: rows 0-7 of C/D ; 1: rows 8-15
    const int l16  = lane & 15;     // column within 16-wide tile
    const int nb   = blockIdx.x * 16;

    // ---- stage X tiles + row constants into LDS
    for (int idx = tid; idx < 3 * 16 * DD; idx += BLK) {
        int mat = idx >> 10;
        int rem = idx & 1023;
        int r = rem >> 6, c = rem & 63;
        const float* src = (mat == 0) ? xi : ((mat == 1) ? xj : xk);
        sX[mat][r][c] = src[(size_t)(nb + r) * DD + c];
    }
    if (tid < 16) {
        sSq[0][tid] = sqxi[nb + tid]; sPn[0][tid] = pnxi[nb + tid];
        sSq[1][tid] = sqxj[nb + tid]; sPn[1][tid] = pnxj[nb + tid];
        sSq[2][tid] = sqxk[nb + tid]; sPn[2][tid] = pnxk[nb + tid];
    }
    __syncthreads();

    // ======================= PHASE 1: row stats ============================
    float m1[8], s1[8], m2[8], s2[8], vmin[8], c2[8];
    int imin[8];
#pragma unroll
    for (int r = 0; r < 8; r++) {
        m1[r] = -1e30f; s1[r] = 0.f;
        m2[r] = -1e30f; s2[r] = 0.f;
        vmin[r] = 1e30f; c2[r] = 0.f; imin[r] = 0x7fffffff;
    }

    for (int ct = wave; ct < MTILES; ct += NWAVES) {
        const int col = ct * 16 + l16;
        const float sqn_c = sqn[col];
        const float pnn_c = pnn[col];
        const float* nrow = nodes + (size_t)col * DD;

        v8f ai = {0.f, 0.f, 0.f, 0.f, 0.f, 0.f, 0.f, 0.f};
        v8f aj = ai, ak = ai;
#pragma unroll
        for (int kk = 0; kk < 16; kk++) {
            const int kc = kk * 4 + half * 2;
            v2f b;  b.x  = nrow[kc];          b.y  = nrow[kc + 1];
            v2f a0; a0.x = sX[0][l16][kc];    a0.y = sX[0][l16][kc + 1];
            ai = __builtin_amdgcn_wmma_f32_16x16x4_f32(false, a0, false, b, (short)0, ai, false, false);
            v2f a1; a1.x = sX[1][l16][kc];    a1.y = sX[1][l16][kc + 1];
            aj = __builtin_amdgcn_wmma_f32_16x16x4_f32(false, a1, false, b, (short)0, aj, false, false);
            v2f a2; a2.x = sX[2][l16][kc];    a2.y = sX[2][l16][kc + 1];
            ak = __builtin_amdgcn_wmma_f32_16x16x4_f32(false, a2, false, b, (short)0, ak, false, false);
        }

#pragma unroll
        for (int r = 0; r < 8; r++) {
            const int rl  = r + half * 8;
            const int row = nb + rl;
            float di = pdistf(ai[r], sSq[0][rl], sPn[0][rl], sqn_c, pnn_c);
            float dj = pdistf(aj[r], sSq[1][rl], sPn[1][rl], sqn_c, pnn_c);
            float dk = pdistf(ak[r], sSq[2][rl], sPn[2][rl], sqn_c, pnn_c);
            // streaming 128MB arrays, read twice total: keep them out of L2 (TH=NT)
            float u1 = __builtin_nontemporal_load(uij  + (size_t)row * MM + col);
            float u2 = __builtin_nontemporal_load(uijk + (size_t)row * MM + col);
            float g1 = __logf(-__logf(u1));
            float g2 = __logf(-__logf(u2));
            float mx = fmaxf(di, dj);
            float z1 = g1 - mx;
            if (z1 > m1[r]) { float sc = __expf(m1[r] - z1); s1[r] = s1[r] * sc + 1.f; m1[r] = z1; }
            else            { s1[r] += __expf(z1 - m1[r]); }
            float mk = fmaxf(dk, mx);
            float z2 = g2 - mk;
            if (z2 > m2[r]) { float sc = __expf(m2[r] - z2); s2[r] = s2[r] * sc + 1.f; c2[r] *= sc; m2[r] = z2; }
            else            { s2[r] += __expf(z2 - m2[r]); }
            if (mx < vmin[r]) { vmin[r] = mx; imin[r] = col; c2[r] = __expf(z2 - m2[r]); }
        }
    }

    // reduce stats across the 16 lanes of each half (columns differ, rows same)
#pragma unroll
    for (int off = 1; off <= 8; off <<= 1) {
#pragma unroll
        for (int r = 0; r < 8; r++) {
            float om1 = __shfl_xor(m1[r], off, 32);
            float os1 = __shfl_xor(s1[r], off, 32);
            float om2 = __shfl_xor(m2[r], off, 32);
            float os2 = __shfl_xor(s2[r], off, 32);
            float ovm = __shfl_xor(vmin[r], off, 32);
            float oc2 = __shfl_xor(c2[r], off, 32);
            int   oim = __shfl_xor(imin[r], off, 32);
            combine_stats(m1[r], s1[r], m2[r], s2[r], vmin[r], c2[r], imin[r],
                          om1, os1, om2, os2, ovm, oc2, oim);
        }
    }
    if (l16 == 0) {
#pragma unroll
        for (int r = 0; r < 8; r++) {
            int rl = r + half * 8;
            redM1[wave][rl] = m1[r]; redS1[wave][rl] = s1[r];
            redM2[wave][rl] = m2[r]; redS2[wave][rl] = s2[r];
            redVM[wave][rl] = vmin[r]; redC2[wave][rl] = c2[r];
            redIM[wave][rl] = imin[r];
        }
    }
    __syncthreads();

    if (tid < 16) {
        float M1 = -1e30f, S1 = 0.f, M2 = -1e30f, S2 = 0.f, VM = 1e30f, C2 = 0.f;
        int IM = 0x7fffffff;
        for (int w = 0; w < NWAVES; w++)
            combine_stats(M1, S1, M2, S2, VM, C2, IM,
                          redM1[w][tid], redS1[w][tid], redM2[w][tid],
                          redS2[w][tid], redVM[w][tid], redC2[w][tid], redIM[w][tid]);
        rowM1[tid] = M1; rowS1[tid] = S1; rowM2[tid] = M2;
        rowZp[tid] = S2 - C2;   // softmax denom with +1000 offset column removed
        rowIm[tid] = IM;
    }
    __syncthreads();

    // ======================= PHASE 2: loss =================================
    float rM1[8], rS1i[8], rM2[8], rZpi[8];
    int   rIm[8];
#pragma unroll
    for (int r = 0; r < 8; r++) {
        int rl = r + half * 8;
        rM1[r] = rowM1[rl]; rS1i[r] = 1.f / rowS1[rl];
        rM2[r] = rowM2[rl]; rZpi[r] = 1.f / rowZp[rl];
        rIm[r] = rowIm[rl];
    }

    float loss = 0.f;
    for (int ct = wave; ct < MTILES; ct += NWAVES) {
        const int col = ct * 16 + l16;
        const float sqn_c = sqn[col];
        const float pnn_c = pnn[col];
        const float* nrow = nodes + (size_t)col * DD;

        v8f ai = {0.f, 0.f, 0.f, 0.f, 0.f, 0.f, 0.f, 0.f};
        v8f aj = ai, ak = ai;
#pragma unroll
        for (int kk = 0; kk < 16; kk++) {
            const int kc = kk * 4 + half * 2;
            v2f b;  b.x  = nrow[kc];          b.y  = nrow[kc + 1];
            v2f a0; a0.x = sX[0][l16][kc];    a0.y = sX[0][l16][kc + 1];
            ai = __builtin_amdgcn_wmma_f32_16x16x4_f32(false, a0, false, b, (short)0, ai, false, false);
            v2f a1; a1.x = sX[1][l16][kc];    a1.y = sX[1][l16][kc + 1];
            aj = __builtin_amdgcn_wmma_f32_16x16x4_f32(false, a1, false, b, (short)0, aj, false, false);
            v2f a2; a2.x = sX[2][l16][kc];    a2.y = sX[2][l16][kc + 1];
            ak = __builtin_amdgcn_wmma_f32_16x16x4_f32(false, a2, false, b, (short)0, ak, false, false);
        }

#pragma unroll
        for (int r = 0; r < 8; r++) {
            const int rl  = r + half * 8;
            const int row = nb + rl;
            float di = pdistf(ai[r], sSq[0][rl], sPn[0][rl], sqn_c, pnn_c);
            float dj = pdistf(aj[r], sSq[1][rl], sPn[1][rl], sqn_c, pnn_c);
            float dk = pdistf(ak[r], sSq[2][rl], sPn[2][rl], sqn_c, pnn_c);
            // last-use streaming reads: non-temporal
            float u1 = __builtin_nontemporal_load(uij  + (size_t)row * MM + col);
            float u2 = __builtin_nontemporal_load(uijk + (size_t)row * MM + col);
            float g1 = __logf(-__logf(u1));
            float g2 = __logf(-__logf(u2));
            float mx = fmaxf(di, dj);
            float mk = fmaxf(dk, mx);
            float lca1 = __expf(g1 - mx - rM1[r]) * rS1i[r];
            float lca2 = (col == rIm[r]) ? 0.f
                        : __expf(g2 - mk - rM2[r]) * rZpi[r];
            float dl = lca1 - lca2;
            loss += softplusf(dl * di) + softplusf(dl * dj) + softplusf(-dl * dk);
        }
    }

    // deterministic block reduction of loss
#pragma unroll
    for (int off = 1; off <= 16; off <<= 1) loss += __shfl_xor(loss, off, 32);
    if (lane == 0) waveLoss[wave] = loss;
    __syncthreads();
    if (tid == 0) {
        float t = 0.f;
        for (int w = 0; w < NWAVES; w++) t += waveLoss[w];
        partials[blockIdx.x] = t;
    }
}

// ------------------------------------------------------- final reduction
__global__ __launch_bounds__(1024) void reduce_partials_kernel(const float* __restrict__ partials,
                                                               float* __restrict__ out) {
    __shared__ float sh[32];
    const int tid = threadIdx.x;
    float v = partials[tid];   // exactly 1024 partials
#pragma unroll
    for (int off = 1; off <= 16; off <<= 1) v += __shfl_xor(v, off, 32);
    if ((tid & 31) == 0) sh[tid >> 5] = v;
    __syncthreads();
    if (tid < 32) {
        float w = sh[tid];
#pragma unroll
        for (int off = 1; off <= 16; off <<= 1) w += __shfl_xor(w, off, 32);
        if (tid == 0) out[0] = w;
    }
}

// ---------------------------------------------------------------- launcher
extern "C" void kernel_launch(void* const* d_in, const int* in_sizes, int n_in,
                              void* d_out, int out_size, void* d_ws, size_t ws_size,
                              hipStream_t stream) {
    (void)in_sizes; (void)n_in; (void)out_size; (void)ws_size;
    const float* xi    = (const float*)d_in[0];
    const float* xj    = (const float*)d_in[1];
    const float* xk    = (const float*)d_in[2];
    const float* nodes = (const float*)d_in[3];
    const float* uij   = (const float*)d_in[4];
    const float* uijk  = (const float*)d_in[5];

    float* ws   = (float*)d_ws;
    float* sqn  = ws;                 // 2048
    float* pnn  = ws + MM;            // 2048
    float* sqxi = ws + 2 * MM;        // 16384 each, six arrays
    float* pnxi = sqxi + NN;
    float* sqxj = pnxi + NN;
    float* pnxj = sqxj + NN;
    float* sqxk = pnxj + NN;
    float* pnxk = sqxk + NN;
    float* partials = pnxk + NN;      // 1024

    rowstats_kernel<<<MM / NWAVES, BLK, 0, stream>>>(nodes, sqn, pnn, MM);
    rowstats_kernel<<<NN / NWAVES, BLK, 0, stream>>>(xi, sqxi, pnxi, NN);
    rowstats_kernel<<<NN / NWAVES, BLK, 0, stream>>>(xj, sqxj, pnxj, NN);
    rowstats_kernel<<<NN / NWAVES, BLK, 0, stream>>>(xk, sqxk, pnxk, NN);

    hhc_main_kernel<<<NN / 16, BLK, 0, stream>>>(
        xi, xj, xk, nodes, uij, uijk,
        sqn, pnn, sqxi, pnxi, sqxj, pnxj, sqxk, pnxk, partials);

    reduce_partials_kernel<<<1, 1024, 0, stream>>>(partials, (float*)d_out);
}